// GATE_75608604279028
// MI455X (gfx1250) — compile-verified
//
#include <hip/hip_runtime.h>

#define NN 50000
#define NE 800000
#define E2T 850000      // NE + NN self loops
#define FIN 128
#define HC 128          // HEADS*HID
#define ED 16
#define OUTC 32
#define NEGS 0.2f

typedef __attribute__((ext_vector_type(2))) float v2f;
typedef __attribute__((ext_vector_type(8))) float v8f;

// ---- helpers ----------------------------------------------------------
__device__ __forceinline__ unsigned fenc(float f) {
  unsigned b = __float_as_uint(f);
  return (b & 0x80000000u) ? ~b : (b | 0x80000000u);
}
__device__ __forceinline__ float fdec(unsigned u) {
  return __uint_as_float((u & 0x80000000u) ? (u & 0x7fffffffu) : ~u);
}
__device__ __forceinline__ float lrelu(float x) { return x > 0.f ? x : NEGS * x; }

// ---- K1: column sums of edge_attr (for self-loop mean) ----------------
__global__ void k_colsum(const float* __restrict__ ea, float* __restrict__ sums) {
  int t = blockIdx.x * blockDim.x + threadIdx.x;
  int c = t & 15;
  int r = t >> 4;
  int stride = (gridDim.x * blockDim.x) >> 4;
  float s = 0.f;
  for (; r < NE; r += stride) s += ea[(size_t)r * ED + c];
  atomicAdd(&sums[c], s);
}

// ---- K2: fold We*a_edge -> M[16x5]; self-loop attention constants -----
// consts: [0..63]=M1[d*4+h], [64..79]=M2[d], [80..83]=a_loop1[h], [84]=a_loop2
__global__ void k_prep(const float* __restrict__ We1, const float* __restrict__ aew1,
                       const float* __restrict__ We2, const float* __restrict__ aew2,
                       const float* __restrict__ sums, float* __restrict__ consts) {
  __shared__ float M1s[64], M2s[16], em[16];
  int t = threadIdx.x;  // 128 threads
  if (t < 16) em[t] = sums[t] * (1.0f / (float)NE);
  if (t < 64) {
    int d = t >> 2, h = t & 3;
    float s = 0.f;
    for (int c = 0; c < 32; ++c) s += We1[d * HC + h * 32 + c] * aew1[h * 32 + c];
    M1s[t] = s;
  }
  if (t >= 64 && t < 80) {
    int d = t - 64;
    float s = 0.f;
    for (int c = 0; c < 32; ++c) s += We2[d * OUTC + c] * aew2[c];
    M2s[d] = s;
  }
  __syncthreads();
  if (t < 64) consts[t] = M1s[t];
  if (t >= 64 && t < 80) consts[t] = M2s[t - 64];
  if (t >= 80 && t < 84) {
    int h = t - 80;
    float s = 0.f;
    for (int d = 0; d < 16; ++d) s += em[d] * M1s[d * 4 + h];
    consts[t] = s;
  }
  if (t == 84) {
    float s = 0.f;
    for (int d = 0; d < 16; ++d) s += em[d] * M2s[d];
    consts[84] = s;
  }
}

// ---- WMMA fp32 GEMM: C[M x NCOLS] = A[M x 128] * B[128 x NCOLS] -------
// grid.x = M/(16*MT); blockDim = 2*NCOLS (one wave per 16-col tile).
// B staged transposed in LDS (padded stride 132 -> conflict-free, const-base
// ds_load_b64 per operand); A strip (16x128) staged per M-tile.
template <int NCOLS, int MT>
__global__ void k_gemm_wmma(const float* __restrict__ A, const float* __restrict__ B,
                            float* __restrict__ C) {
  constexpr int LDP = 132;  // padded LDS row stride in floats
  __shared__ float As[16 * LDP];
  __shared__ float BTs[NCOLS * LDP];
  const int tid = threadIdx.x;
  constexpr int NT = 2 * NCOLS;   // threads per block
  constexpr int NQ = NCOLS / 4;   // float4 per B row
  for (int j = tid; j < 128 * NQ; j += NT) {
    int k = j / NQ, nq = j % NQ;
    float4 v = *(const float4*)&B[(size_t)k * NCOLS + nq * 4];
    BTs[(nq * 4 + 0) * LDP + k] = v.x;
    BTs[(nq * 4 + 1) * LDP + k] = v.y;
    BTs[(nq * 4 + 2) * LDP + k] = v.z;
    BTs[(nq * 4 + 3) * LDP + k] = v.w;
  }
  const int lane = tid & 31, wave = tid >> 5;
  const int r = lane & 15;           // A row / B col / D col within tile
  const int kk = (lane >> 4) << 1;   // lanes 16-31 carry K+2,K+3
  const int cn = wave * 16 + r;
  const float* aptr = &As[r * LDP + kk];
  const float* bptr = &BTs[cn * LDP + kk];
  for (int s = 0; s < MT; ++s) {
    const int tileM = (blockIdx.x * MT + s) * 16;
    __syncthreads();  // BTs ready (s==0) / prior strip's compute done (s>0)
    const float4* Av = (const float4*)(A + (size_t)tileM * 128);
    for (int i = tid; i < 512; i += NT) {
      float4 v = Av[i];
      *(float4*)&As[(i >> 5) * LDP + (i & 31) * 4] = v;
    }
    __syncthreads();
    v8f acc = {};
#pragma unroll
    for (int k0 = 0; k0 < 128; k0 += 4) {
      v2f a = *(const v2f*)&aptr[k0];
      v2f b = *(const v2f*)&bptr[k0];
      acc = __builtin_amdgcn_wmma_f32_16x16x4_f32(false, a, false, b, (short)0, acc,
                                                  false, false);
    }
    const int m0 = tileM + ((lane >> 4) << 3);
#pragma unroll
    for (int i = 0; i < 8; ++i) C[(size_t)(m0 + i) * NCOLS + cn] = acc[i];
  }
}

// ---- K4: per-node attention dots, layer 1 (wave per node) -------------
__global__ void k_nodeproj1(const float* __restrict__ h1, const float* __restrict__ aw,
                            const float* __restrict__ dw, float* __restrict__ as1,
                            float* __restrict__ ad1) {
  int wave = threadIdx.x >> 5, lane = threadIdx.x & 31;
  int n = blockIdx.x * 8 + wave;
  if (n >= NN) return;
  float4 h = *(const float4*)&h1[(size_t)n * HC + lane * 4];
  float4 a = *(const float4*)&aw[lane * 4];
  float4 d = *(const float4*)&dw[lane * 4];
  float ps = h.x * a.x + h.y * a.y + h.z * a.z + h.w * a.w;
  float pd = h.x * d.x + h.y * d.y + h.z * d.z + h.w * d.w;
  ps += __shfl_xor(ps, 1); pd += __shfl_xor(pd, 1);
  ps += __shfl_xor(ps, 2); pd += __shfl_xor(pd, 2);
  ps += __shfl_xor(ps, 4); pd += __shfl_xor(pd, 4);
  if ((lane & 7) == 0) {
    int hh = lane >> 3;
    as1[n * 4 + hh] = ps;
    ad1[n * 4 + hh] = pd;
  }
}

// ---- K5: a_edge for both layers in one pass: e_attr @ M[16x5] ---------
__global__ void k_edgeproj(const float* __restrict__ ea, const float* __restrict__ consts,
                           float* __restrict__ ae1, float* __restrict__ ae2) {
  __shared__ float M[80];
  if (threadIdx.x < 80) M[threadIdx.x] = consts[threadIdx.x];
  __syncthreads();
  int e = blockIdx.x * blockDim.x + threadIdx.x;
  if (e >= NE) return;
  const float* row = ea + (size_t)e * ED;
  float a0 = 0.f, a1 = 0.f, a2 = 0.f, a3 = 0.f, a4 = 0.f;
#pragma unroll
  for (int d = 0; d < 16; ++d) {
    float v = row[d];
    a0 += v * M[d * 4 + 0]; a1 += v * M[d * 4 + 1];
    a2 += v * M[d * 4 + 2]; a3 += v * M[d * 4 + 3];
    a4 += v * M[64 + d];
  }
  float4 o = {a0, a1, a2, a3};
  *(float4*)&ae1[(size_t)e * 4] = o;
  ae2[e] = a4;
}

// ---- layer-1 softmax passes ------------------------------------------
__device__ __forceinline__ void edge1_sd(int e, int h, const int* idx, const float* ae1,
                                         const float* consts, int& s, int& d, float& ae) {
  if (e < NE) { s = idx[e]; d = idx[NE + e]; ae = ae1[e * 4 + h]; }
  else        { s = d = e - NE; ae = consts[80 + h]; }
}

__global__ void k_max1(const int* __restrict__ idx, const float* __restrict__ as1,
                       const float* __restrict__ ad1, const float* __restrict__ ae1,
                       const float* __restrict__ consts, unsigned* __restrict__ m1) {
  int t = blockIdx.x * blockDim.x + threadIdx.x;
  if (t >= E2T * 4) return;
  int e = t >> 2, h = t & 3, s, d; float ae;
  edge1_sd(e, h, idx, ae1, consts, s, d, ae);
  float lg = lrelu(as1[s * 4 + h] + ad1[d * 4 + h] + ae);
  atomicMax(&m1[d * 4 + h], fenc(lg));
}

__global__ void k_exp1(const int* __restrict__ idx, const float* __restrict__ as1,
                       const float* __restrict__ ad1, const float* __restrict__ ae1,
                       const float* __restrict__ consts, const unsigned* __restrict__ m1,
                       float* __restrict__ ex1, float* __restrict__ den1) {
  int t = blockIdx.x * blockDim.x + threadIdx.x;
  if (t >= E2T * 4) return;
  int e = t >> 2, h = t & 3, s, d; float ae;
  edge1_sd(e, h, idx, ae1, consts, s, d, ae);
  float lg = lrelu(as1[s * 4 + h] + ad1[d * 4 + h] + ae);
  float ex = __expf(lg - fdec(m1[d * 4 + h]));
  ex1[t] = ex;
  atomicAdd(&den1[d * 4 + h], ex);
}

// one thread per (edge, float4-of-channels): E2T*32 threads
__global__ void k_agg1(const int* __restrict__ idx, const float* __restrict__ h1,
                       const float* __restrict__ ex1, const float* __restrict__ den1,
                       float* __restrict__ out1) {
  int t = blockIdx.x * blockDim.x + threadIdx.x;
  if (t >= E2T * 32) return;
  int e = t >> 5, q = t & 31, h = q >> 3;
  int s, d;
  if (e < NE) { s = idx[e]; d = idx[NE + e]; } else { s = d = e - NE; }
  float alpha = ex1[e * 4 + h] / (den1[d * 4 + h] + 1e-16f);
  float4 hv = *(const float4*)&h1[(size_t)s * HC + q * 4];
  float* o = &out1[(size_t)d * HC + q * 4];
  atomicAdd(&o[0], alpha * hv.x);
  atomicAdd(&o[1], alpha * hv.y);
  atomicAdd(&o[2], alpha * hv.z);
  atomicAdd(&o[3], alpha * hv.w);
}

__global__ void k_relubias(float* __restrict__ h2, const float* __restrict__ b1) {
  int t = blockIdx.x * blockDim.x + threadIdx.x;
  if (t >= NN * HC) return;
  float v = h2[t] + b1[t & 127];
  h2[t] = v > 0.f ? v : 0.f;
}

// ---- layer 2 ----------------------------------------------------------
__global__ void k_nodeproj2(const float* __restrict__ hh, const float* __restrict__ aw,
                            const float* __restrict__ dw, float* __restrict__ as2,
                            float* __restrict__ ad2) {
  int wave = threadIdx.x >> 5, lane = threadIdx.x & 31;
  int n = blockIdx.x * 8 + wave;
  if (n >= NN) return;
  float v = hh[(size_t)n * OUTC + lane];
  float ps = v * aw[lane], pd = v * dw[lane];
  for (int o = 1; o < 32; o <<= 1) { ps += __shfl_xor(ps, o); pd += __shfl_xor(pd, o); }
  if (lane == 0) { as2[n] = ps; ad2[n] = pd; }
}

__global__ void k_max2(const int* __restrict__ idx, const float* __restrict__ as2,
                       const float* __restrict__ ad2, const float* __restrict__ ae2,
                       const float* __restrict__ consts, unsigned* __restrict__ m2) {
  int e = blockIdx.x * blockDim.x + threadIdx.x;
  if (e >= E2T) return;
  int s, d; float ae;
  if (e < NE) { s = idx[e]; d = idx[NE + e]; ae = ae2[e]; }
  else        { s = d = e - NE; ae = consts[84]; }
  float lg = lrelu(as2[s] + ad2[d] + ae);
  atomicMax(&m2[d], fenc(lg));
}

__global__ void k_exp2(const int* __restrict__ idx, const float* __restrict__ as2,
                       const float* __restrict__ ad2, const float* __restrict__ ae2,
                       const float* __restrict__ consts, const unsigned* __restrict__ m2,
                       float* __restrict__ ex2, float* __restrict__ den2) {
  int e = blockIdx.x * blockDim.x + threadIdx.x;
  if (e >= E2T) return;
  int s, d; float ae;
  if (e < NE) { s = idx[e]; d = idx[NE + e]; ae = ae2[e]; }
  else        { s = d = e - NE; ae = consts[84]; }
  float lg = lrelu(as2[s] + ad2[d] + ae);
  float ex = __expf(lg - fdec(m2[d]));
  ex2[e] = ex;
  atomicAdd(&den2[d], ex);
}

// one thread per (edge, float4-of-channels): E2T*8 threads
__global__ void k_agg2(const int* __restrict__ idx, const float* __restrict__ hh,
                       const float* __restrict__ ex2, const float* __restrict__ den2,
                       float* __restrict__ out) {
  int t = blockIdx.x * blockDim.x + threadIdx.x;
  if (t >= E2T * 8) return;
  int e = t >> 3, q = t & 7;
  int s, d;
  if (e < NE) { s = idx[e]; d = idx[NE + e]; } else { s = d = e - NE; }
  float alpha = ex2[e] / (den2[d] + 1e-16f);
  float4 hv = *(const float4*)&hh[(size_t)s * OUTC + q * 4];
  float* o = &out[(size_t)d * OUTC + q * 4];
  atomicAdd(&o[0], alpha * hv.x);
  atomicAdd(&o[1], alpha * hv.y);
  atomicAdd(&o[2], alpha * hv.z);
  atomicAdd(&o[3], alpha * hv.w);
}

__global__ void k_bias2(float* __restrict__ out, const float* __restrict__ b2) {
  int t = blockIdx.x * blockDim.x + threadIdx.x;
  if (t >= NN * OUTC) return;
  out[t] += b2[t & 31];
}

// ---- host -------------------------------------------------------------
extern "C" void kernel_launch(void* const* d_in, const int* in_sizes, int n_in,
                              void* d_out_v, int out_size, void* d_ws, size_t ws_size,
                              hipStream_t stream) {
  const float* x    = (const float*)d_in[0];
  const int*   idx  = (const int*)d_in[1];
  const float* ea   = (const float*)d_in[2];
  const float* W1   = (const float*)d_in[3];
  const float* We1  = (const float*)d_in[4];
  const float* asw1 = (const float*)d_in[5];
  const float* adw1 = (const float*)d_in[6];
  const float* aew1 = (const float*)d_in[7];
  const float* b1   = (const float*)d_in[8];
  const float* W2   = (const float*)d_in[9];
  const float* We2  = (const float*)d_in[10];
  const float* asw2 = (const float*)d_in[11];
  const float* adw2 = (const float*)d_in[12];
  const float* aew2 = (const float*)d_in[13];
  const float* b2   = (const float*)d_in[14];
  float* out = (float*)d_out_v;

  size_t cur = 0;
  char* base = (char*)d_ws;
  auto alloc = [&](size_t bytes) -> char* {
    char* p = base + cur;
    cur = (cur + bytes + 255) & ~(size_t)255;
    return p;
  };
  float*    sums   = (float*)alloc(16 * sizeof(float));
  float*    consts = (float*)alloc(96 * sizeof(float));
  float*    h1     = (float*)alloc((size_t)NN * HC * sizeof(float));
  float*    as1    = (float*)alloc((size_t)NN * 4 * sizeof(float));
  float*    ad1    = (float*)alloc((size_t)NN * 4 * sizeof(float));
  float*    ae1    = (float*)alloc((size_t)NE * 4 * sizeof(float));
  float*    ae2    = (float*)alloc((size_t)NE * sizeof(float));
  unsigned* m1     = (unsigned*)alloc((size_t)NN * 4 * sizeof(unsigned));
  float*    den1   = (float*)alloc((size_t)NN * 4 * sizeof(float));
  float*    ex1    = (float*)alloc((size_t)E2T * 4 * sizeof(float));
  float*    h2     = (float*)alloc((size_t)NN * HC * sizeof(float));   // agg1 out, relu in place
  float*    hh     = (float*)alloc((size_t)NN * OUTC * sizeof(float));
  float*    as2    = (float*)alloc((size_t)NN * sizeof(float));
  float*    ad2    = (float*)alloc((size_t)NN * sizeof(float));
  unsigned* m2     = (unsigned*)alloc((size_t)NN * sizeof(unsigned));
  float*    den2   = (float*)alloc((size_t)NN * sizeof(float));
  float*    ex2    = (float*)alloc((size_t)E2T * sizeof(float));
  (void)ws_size; (void)in_sizes; (void)n_in; (void)out_size;

  hipMemsetAsync(sums, 0, 16 * sizeof(float), stream);
  hipMemsetAsync(m1, 0, (size_t)NN * 4 * sizeof(unsigned), stream);
  hipMemsetAsync(den1, 0, (size_t)NN * 4 * sizeof(float), stream);
  hipMemsetAsync(h2, 0, (size_t)NN * HC * sizeof(float), stream);
  hipMemsetAsync(m2, 0, (size_t)NN * sizeof(unsigned), stream);
  hipMemsetAsync(den2, 0, (size_t)NN * sizeof(float), stream);
  hipMemsetAsync(out, 0, (size_t)NN * OUTC * sizeof(float), stream);

  k_colsum<<<512, 256, 0, stream>>>(ea, sums);
  k_prep<<<1, 128, 0, stream>>>(We1, aew1, We2, aew2, sums, consts);

  // h1 = x@W1 : 3125 M-tiles = 625 blocks x 5 strips
  k_gemm_wmma<128, 5><<<625, 256, 0, stream>>>(x, W1, h1);
  k_nodeproj1<<<NN / 8, 256, 0, stream>>>(h1, asw1, adw1, as1, ad1);
  k_edgeproj<<<(NE + 255) / 256, 256, 0, stream>>>(ea, consts, ae1, ae2);

  k_max1<<<(E2T * 4 + 255) / 256, 256, 0, stream>>>(idx, as1, ad1, ae1, consts, m1);
  k_exp1<<<(E2T * 4 + 255) / 256, 256, 0, stream>>>(idx, as1, ad1, ae1, consts, m1, ex1, den1);
  k_agg1<<<(E2T * 32) / 256, 256, 0, stream>>>(idx, h1, ex1, den1, h2);
  k_relubias<<<(NN * HC + 255) / 256, 256, 0, stream>>>(h2, b1);

  // hh = h2@W2
  k_gemm_wmma<32, 5><<<625, 64, 0, stream>>>(h2, W2, hh);
  k_nodeproj2<<<NN / 8, 256, 0, stream>>>(hh, asw2, adw2, as2, ad2);

  k_max2<<<(E2T + 255) / 256, 256, 0, stream>>>(idx, as2, ad2, ae2, consts, m2);
  k_exp2<<<(E2T + 255) / 256, 256, 0, stream>>>(idx, as2, ad2, ae2, consts, m2, ex2, den2);
  k_agg2<<<(E2T * 8 + 255) / 256, 256, 0, stream>>>(idx, hh, ex2, den2, out);
  k_bias2<<<(NN * OUTC + 255) / 256, 256, 0, stream>>>(out, b2);
}